// MultiScaleFrequencyLoss_79465484910838
// MI455X (gfx1250) — compile-verified
//
#include <hip/hip_runtime.h>
#include <math.h>

// ---------------------------------------------------------------------------
// MultiScaleFrequencyLoss on MI455X (gfx1250, wave32)
//
// 2-D FFT computed as two dense GEMM passes against the N x N complex DFT
// matrix, executed with V_WMMA_F32_16X16X4_F32 (fp32 matrix pipe, full
// precision). Each wave computes a 32x32 output region (2x2 WMMA tiles).
// N is a compile-time template parameter (512/256/128) so trip counts and
// strides fold into immediates and no spill-prone generic loop paths remain.
// Stage-2 epilogue is fused with |F|, log, radial-PSD binning and the
// fourier-loss reduction so the complex spectrum never touches HBM.
// ---------------------------------------------------------------------------

typedef float v2f __attribute__((ext_vector_type(2)));
typedef float v8f __attribute__((ext_vector_type(8)));

#define MSFL_EPS 1e-8f
#define MSFL_PI  3.14159265358979323846f

// ------------------------------- small kernels ------------------------------

__global__ void msfl_fill_dft(float* __restrict__ Dr, float* __restrict__ Di, int N) {
    long idx = (long)blockIdx.x * blockDim.x + threadIdx.x;
    long total = (long)N * N;
    if (idx >= total) return;
    int v = (int)(idx % N);
    int x = (int)(idx / N);
    int ph = (int)(((long)x * (long)v) % (long)N);      // exact phase mod N
    float ang = -2.0f * MSFL_PI * (float)ph / (float)N;
    float s, c;
    __sincosf(ang, &s, &c);
    Dr[idx] = c;
    Di[idx] = s;
}

__global__ void msfl_avg_pool(const float* __restrict__ src, float* __restrict__ dst,
                              int H, int W, int s, int G) {
    int h2 = H / s, w2 = W / s;
    long total = (long)G * h2 * w2;
    long idx = (long)blockIdx.x * blockDim.x + threadIdx.x;
    if (idx >= total) return;
    int x = (int)(idx % w2);
    long t = idx / w2;
    int y = (int)(t % h2);
    int g = (int)(t / h2);
    const float* sg = src + (long)g * H * W;
    float acc = 0.0f;
    for (int dy = 0; dy < s; ++dy)
        for (int dx = 0; dx < s; ++dx)
            acc += sg[(long)(y * s + dy) * W + (x * s + dx)];
    dst[idx] = acc * (1.0f / (float)(s * s));
}

__global__ void msfl_radial_counts(float* __restrict__ counts, int N, int max_r) {
    long idx = (long)blockIdx.x * blockDim.x + threadIdx.x;
    long total = (long)N * N;
    if (idx >= total) return;
    int v = (int)(idx % N);
    int u = (int)(idx / N);
    float dy = (float)(u - N / 2);
    float dx = (float)(v - N / 2);
    int bin = (int)floorf(sqrtf(dy * dy + dx * dx));
    if (bin > max_r) bin = max_r;
    atomicAdd(&counts[bin], 1.0f);
}

__global__ void msfl_radial_finalize(const float* __restrict__ sp,
                                     const float* __restrict__ st,
                                     const float* __restrict__ counts,
                                     int max_r, int nbins,
                                     float* __restrict__ ssum) {
    __shared__ float red[256];
    int g = blockIdx.x;
    float local = 0.0f;
    for (int r = threadIdx.x; r < max_r; r += blockDim.x) {
        float c = fmaxf(counts[r], 1.0f);
        float ap = sp[(long)g * nbins + r] / c;
        float at = st[(long)g * nbins + r] / c;
        local += fabsf(ap - at);
    }
    red[threadIdx.x] = local;
    __syncthreads();
    for (int s = 128; s > 0; s >>= 1) {
        if (threadIdx.x < s) red[threadIdx.x] += red[threadIdx.x + s];
        __syncthreads();
    }
    if (threadIdx.x == 0) atomicAdd(ssum, red[0]);
}

__global__ void msfl_scale_combine(float* __restrict__ out,
                                   const float* __restrict__ acc,  // [0]=fourier sum, [1]=spectral sum
                                   float inv_s, float fnorm, float snorm) {
    if (threadIdx.x == 0 && blockIdx.x == 0)
        out[0] += inv_s * (1.0f * acc[0] * fnorm + 0.5f * acc[1] * snorm);
}

// ------------------------------ WMMA DFT stages -----------------------------
// Fragment layouts for V_WMMA_F32_16X16X4_F32 (wave32):
//   A (16x4):  lane l holds A[l%16, 2*(l/16)+j] in vgpr j (j=0,1)
//   B (4x16):  lane l holds B[2*(l/16)+j, l%16] in vgpr j
//   C/D 16x16: vgpr i, lane l -> (M = i + 8*(l/16), N = l%16)
//
// Each wave computes a 32x32 output region: sub-tiles (mi, nj), mi,nj in {0,1}.

// Stage 1: T = X * D  (real X times complex symmetric DFT matrix)
template <int N>
__global__ void msfl_dft_stage1(const float* __restrict__ X,
                                const float* __restrict__ Dr,
                                const float* __restrict__ Di,
                                float* __restrict__ Tr,
                                float* __restrict__ Ti) {
    const int lane = threadIdx.x;
    const int lm = lane & 15;
    const int lh = lane >> 4;            // 0 or 1
    const int n0 = blockIdx.x * 32;      // output column region (v)
    const int m0 = blockIdx.y * 32;      // output row region (y)
    const int g = blockIdx.z;
    const float* Xg = X + (long)g * N * N;

    // per-wave invariant bases (int offsets; N*N <= 2^18, g*N*N < 2^24)
    const float* a0 = Xg + (m0 + lm) * N + 2 * lh;      // + 16*N for sub-tile 1
    const int bcol0 = 2 * lh * N + n0 + lm;             // + 16 for sub-tile 1

    v8f accR[2][2] = {};
    v8f accI[2][2] = {};
    for (int k0 = 0; k0 < N; k0 += 4) {
        v2f a[2], br[2], bm[2];
        #pragma unroll
        for (int i = 0; i < 2; ++i)
            a[i] = *(const v2f*)(a0 + i * 16 * N + k0);
        #pragma unroll
        for (int j = 0; j < 2; ++j) {
            int bi = bcol0 + j * 16 + k0 * N;
            br[j].x = Dr[bi]; br[j].y = Dr[bi + N];
            bm[j].x = Di[bi]; bm[j].y = Di[bi + N];
        }
        #pragma unroll
        for (int i = 0; i < 2; ++i)
            #pragma unroll
            for (int j = 0; j < 2; ++j) {
                accR[i][j] = __builtin_amdgcn_wmma_f32_16x16x4_f32(false, a[i], false, br[j], (short)0, accR[i][j], false, false);
                accI[i][j] = __builtin_amdgcn_wmma_f32_16x16x4_f32(false, a[i], false, bm[j], (short)0, accI[i][j], false, false);
            }
    }
    float* trg = Tr + (long)g * N * N;
    float* tig = Ti + (long)g * N * N;
    #pragma unroll
    for (int i = 0; i < 2; ++i)
        #pragma unroll
        for (int j = 0; j < 2; ++j)
            #pragma unroll
            for (int e = 0; e < 8; ++e) {
                int m = m0 + 16 * i + e + 8 * lh;
                int n = n0 + 16 * j + lm;
                trg[m * N + n] = accR[i][j][e];
                tig[m * N + n] = accI[i][j][e];
            }
}

// Stage 2: F = D * T (complex x complex), fused loss epilogue.
// MODE 0: predicted  -> write logmag to LM, accumulate PSD radial sums.
// MODE 1: target     -> fourier |logmag_p - logmag_t| into fsum, PSD radial sums.
template <int N, int MODE>
__global__ void msfl_dft_stage2(const float* __restrict__ Tr,
                                const float* __restrict__ Ti,
                                const float* __restrict__ Dr,
                                const float* __restrict__ Di,
                                float* __restrict__ LM,
                                float* __restrict__ rsums,
                                float* __restrict__ fsum) {
    constexpr int max_r = N / 2;
    constexpr int nbins = max_r + 1;
    const int lane = threadIdx.x;
    const int lm = lane & 15;
    const int lh = lane >> 4;
    const int n0 = blockIdx.x * 32;      // v
    const int m0 = blockIdx.y * 32;      // u
    const int g = blockIdx.z;
    const float* trg = Tr + (long)g * N * N;
    const float* tig = Ti + (long)g * N * N;

    const int arow0 = (m0 + lm) * N + 2 * lh;           // + 16*N for sub-tile 1
    const int bcol0 = 2 * lh * N + n0 + lm;             // + 16 for sub-tile 1

    v8f accFr[2][2] = {};
    v8f accFi[2][2] = {};
    for (int k0 = 0; k0 < N; k0 += 4) {
        v2f aR[2], aI[2], aIn[2], bR[2], bI[2];
        #pragma unroll
        for (int i = 0; i < 2; ++i) {
            int ai = arow0 + i * 16 * N + k0;
            aR[i] = *(const v2f*)(Dr + ai);
            aI[i] = *(const v2f*)(Di + ai);
            aIn[i].x = -aI[i].x; aIn[i].y = -aI[i].y;   // -Di (f32 WMMA has no A-neg)
        }
        #pragma unroll
        for (int j = 0; j < 2; ++j) {
            int bi = bcol0 + j * 16 + k0 * N;
            bR[j].x = trg[bi]; bR[j].y = trg[bi + N];
            bI[j].x = tig[bi]; bI[j].y = tig[bi + N];
        }
        #pragma unroll
        for (int i = 0; i < 2; ++i)
            #pragma unroll
            for (int j = 0; j < 2; ++j) {
                // Fr += Dr*Tr - Di*Ti ; Fi += Dr*Ti + Di*Tr
                accFr[i][j] = __builtin_amdgcn_wmma_f32_16x16x4_f32(false, aR[i],  false, bR[j], (short)0, accFr[i][j], false, false);
                accFr[i][j] = __builtin_amdgcn_wmma_f32_16x16x4_f32(false, aIn[i], false, bI[j], (short)0, accFr[i][j], false, false);
                accFi[i][j] = __builtin_amdgcn_wmma_f32_16x16x4_f32(false, aR[i],  false, bI[j], (short)0, accFi[i][j], false, false);
                accFi[i][j] = __builtin_amdgcn_wmma_f32_16x16x4_f32(false, aI[i],  false, bR[j], (short)0, accFi[i][j], false, false);
            }
    }

    float* lmg = LM + (long)g * N * N;
    float* rsg = rsums + g * nbins;
    float lsum = 0.0f;
    #pragma unroll
    for (int i = 0; i < 2; ++i)
        #pragma unroll
        for (int j = 0; j < 2; ++j)
            #pragma unroll
            for (int e = 0; e < 8; ++e) {
                int u = m0 + 16 * i + e + 8 * lh;
                int v = n0 + 16 * j + lm;
                float fr = accFr[i][j][e], fi = accFi[i][j][e];
                float mag2 = fr * fr + fi * fi;
                float lmag = __logf(sqrtf(mag2) + MSFL_EPS);
                float psd  = __logf(mag2 + MSFL_EPS);
                float dy = (float)(u - N / 2);
                float dx = (float)(v - N / 2);
                int bin = (int)floorf(sqrtf(dy * dy + dx * dx));
                if (bin > max_r) bin = max_r;
                atomicAdd(&rsg[bin], psd);
                if (MODE == 0) {
                    lmg[u * N + v] = lmag;
                } else {
                    lsum += fabsf(lmg[u * N + v] - lmag);
                }
            }
    if (MODE == 1) {
        #pragma unroll
        for (int o = 16; o > 0; o >>= 1)
            lsum += __shfl_xor(lsum, o, 32);
        if (lane == 0) atomicAdd(fsum, lsum);
    }
}

// ------------------------- per-scale GEMM dispatch ---------------------------

template <int N>
static void msfl_run_scale_gemms(const float* Xp, const float* Xt,
                                 float* Dr, float* Di, float* Tr, float* Ti,
                                 float* LM, float* sumsP, float* sumsT, float* acc,
                                 int G, hipStream_t stream) {
    dim3 tg(N / 32, N / 32, G);
    msfl_dft_stage1<N><<<tg, 32, 0, stream>>>(Xp, Dr, Di, Tr, Ti);
    msfl_dft_stage2<N, 0><<<tg, 32, 0, stream>>>(Tr, Ti, Dr, Di, LM, sumsP, acc);
    msfl_dft_stage1<N><<<tg, 32, 0, stream>>>(Xt, Dr, Di, Tr, Ti);
    msfl_dft_stage2<N, 1><<<tg, 32, 0, stream>>>(Tr, Ti, Dr, Di, LM, sumsT, acc);
}

// --------------------------------- launcher ---------------------------------

extern "C" void kernel_launch(void* const* d_in, const int* in_sizes, int n_in,
                              void* d_out, int out_size, void* d_ws, size_t ws_size,
                              hipStream_t stream) {
    (void)in_sizes; (void)n_in; (void)ws_size;
    const float* pred = (const float*)d_in[0];
    const float* targ = (const float*)d_in[1];
    float* out = (float*)d_out;

    const int G = 16 * 3;          // flattened (b, c) planes
    const int H0 = 512;

    // workspace carve-up (256-byte aligned regions)
    char* ws = (char*)d_ws;
    size_t off = 0;
    auto carve = [&](size_t bytes) -> float* {
        float* p = (float*)(ws + off);
        off += (bytes + 255) & ~(size_t)255;
        return p;
    };
    float* Dr     = carve((size_t)512 * 512 * 4);
    float* Di     = carve((size_t)512 * 512 * 4);
    float* Tr     = carve((size_t)G * 512 * 512 * 4);
    float* Ti     = carve((size_t)G * 512 * 512 * 4);
    float* LM     = carve((size_t)G * 512 * 512 * 4);
    float* poolP  = carve((size_t)G * 256 * 256 * 4);
    float* poolT  = carve((size_t)G * 256 * 256 * 4);
    float* sumsP  = carve((size_t)G * 257 * 4);
    float* sumsT  = carve((size_t)G * 257 * 4);
    float* counts = carve((size_t)257 * 4);
    float* acc    = carve((size_t)2 * 4);      // [0]=fourier sum, [1]=spectral sum

    hipMemsetAsync(d_out, 0, (size_t)out_size * sizeof(float), stream);

    const int scales[3] = {1, 2, 4};
    for (int si = 0; si < 3; ++si) {
        const int s = scales[si];
        const int N = H0 / s;
        const int max_r = N / 2;
        const int nbins = max_r + 1;

        const float* Xp;
        const float* Xt;
        if (s == 1) {
            Xp = pred;
            Xt = targ;
        } else {
            long total = (long)G * (H0 / s) * (H0 / s);
            int blk = (int)((total + 255) / 256);
            msfl_avg_pool<<<blk, 256, 0, stream>>>(pred, poolP, H0, H0, s, G);
            msfl_avg_pool<<<blk, 256, 0, stream>>>(targ, poolT, H0, H0, s, G);
            Xp = poolP;
            Xt = poolT;
        }

        {
            long total = (long)N * N;
            int blk = (int)((total + 255) / 256);
            msfl_fill_dft<<<blk, 256, 0, stream>>>(Dr, Di, N);
        }

        hipMemsetAsync(sumsP, 0, (size_t)G * nbins * 4, stream);
        hipMemsetAsync(sumsT, 0, (size_t)G * nbins * 4, stream);
        hipMemsetAsync(counts, 0, (size_t)nbins * 4, stream);
        hipMemsetAsync(acc, 0, 2 * 4, stream);

        {
            long total = (long)N * N;
            int blk = (int)((total + 255) / 256);
            msfl_radial_counts<<<blk, 256, 0, stream>>>(counts, N, max_r);
        }

        if (N == 512)      msfl_run_scale_gemms<512>(Xp, Xt, Dr, Di, Tr, Ti, LM, sumsP, sumsT, acc, G, stream);
        else if (N == 256) msfl_run_scale_gemms<256>(Xp, Xt, Dr, Di, Tr, Ti, LM, sumsP, sumsT, acc, G, stream);
        else               msfl_run_scale_gemms<128>(Xp, Xt, Dr, Di, Tr, Ti, LM, sumsP, sumsT, acc, G, stream);

        msfl_radial_finalize<<<G, 256, 0, stream>>>(sumsP, sumsT, counts, max_r, nbins, &acc[1]);

        float fnorm = 1.0f / ((float)G * (float)N * (float)N);
        float snorm = 1.0f / ((float)G * (float)max_r);
        msfl_scale_combine<<<1, 1, 0, stream>>>(out, acc, 1.0f / (float)s, fnorm, snorm);
    }
}